// m1_2190433321036
// MI455X (gfx1250) — compile-verified
//
#include <hip/hip_runtime.h>

// CDNA5 / gfx1250: wave32, WMMA (not MFMA).
typedef __attribute__((ext_vector_type(2))) float v2f;
typedef __attribute__((ext_vector_type(4))) float v4f;
typedef __attribute__((ext_vector_type(8))) float v8f;

#define T_LEN      (1 << 24)   // 16,777,216
#define NBLK       4096
#define BLK        256
#define PER_THREAD 16          // NBLK * BLK * PER_THREAD == T_LEN

// Ordered geometric-weighted combine of 256 partials using fp32 WMMA.
// part[i] are segment B-values in time order; each segment has gain p.
// Returns h = sum_i p^(255-i) * part[i]  (exact fp32 via V_WMMA_F32_16X16X4_F32).
// Must be called by threads 0..31 of a block (one full wave32, EXEC all-1s).
// NO register arrays: all powers are named scalars, so the lane-half choice is
// a single v_cndmask between two uniform SSA values (no index-select ladders).
__device__ __forceinline__ float geo_combine256(const float* part, float p) {
    const int  lane = threadIdx.x & 31;
    const bool hi   = lane >= 16;
    const int  row  = lane & 15;

    // p^0 .. p^15 as named scalars (uniform across lanes)
    const float p0 = 1.0f;
    const float p1 = p;
    const float p2  = p1  * p;
    const float p3  = p2  * p;
    const float p4  = p3  * p;
    const float p5  = p4  * p;
    const float p6  = p5  * p;
    const float p7  = p6  * p;
    const float p8  = p7  * p;
    const float p9  = p8  * p;
    const float p10 = p9  * p;
    const float p11 = p10 * p;
    const float p12 = p11 * p;
    const float p13 = p12 * p;
    const float p14 = p13 * p;
    const float p15 = p14 * p;
    const float qq  = p15 * p;   // q = p^16

    // q^0 .. q^15 as named scalars
    const float q0 = 1.0f;
    const float q1 = qq;
    const float q2  = q1  * qq;
    const float q3  = q2  * qq;
    const float q4  = q3  * qq;
    const float q5  = q4  * qq;
    const float q6  = q5  * qq;
    const float q7  = q6  * qq;
    const float q8  = q7  * qq;
    const float q9  = q8  * qq;
    const float q10 = q9  * qq;
    const float q11 = q10 * qq;
    const float q12 = q11 * qq;
    const float q13 = q12 * qq;
    const float q14 = q13 * qq;
    const float q15 = q14 * qq;

    // Stage 1: s_r = sum_c p^(15-c) * V[r][c],  V[r][c] = part[16r + c].
    // A layout (16x4 f32): lanes 0-15 hold K=klo,klo+1; lanes 16-31 hold K=klo+2,klo+3.
    const int abase = row * 16 + (hi ? 2 : 0);
    v2f A0, A1, A2, A3;
    A0.x = part[abase + 0];   A0.y = part[abase + 1];
    A1.x = part[abase + 4];   A1.y = part[abase + 5];
    A2.x = part[abase + 8];   A2.y = part[abase + 9];
    A3.x = part[abase + 12];  A3.y = part[abase + 13];

    // B layout (4x16 f32): VGPR0 = rows k=0 (lo half) / k=2 (hi half); VGPR1 = k=1 / k=3.
    // Row k of chunk kb carries weight p^(15-(4kb+k)), replicated over all 16 columns.
    v2f B0, B1, B2, B3;
    B0.x = hi ? p13 : p15;    B0.y = hi ? p12 : p14;
    B1.x = hi ? p9  : p11;    B1.y = hi ? p8  : p10;
    B2.x = hi ? p5  : p7;     B2.y = hi ? p4  : p6;
    B3.x = hi ? p1  : p3;     B3.y = hi ? p0  : p2;

    v8f acc = {};
    acc = __builtin_amdgcn_wmma_f32_16x16x4_f32(false, A0, false, B0, (short)0, acc, false, false);
    acc = __builtin_amdgcn_wmma_f32_16x16x4_f32(false, A1, false, B1, (short)0, acc, false, false);
    acc = __builtin_amdgcn_wmma_f32_16x16x4_f32(false, A2, false, B2, (short)0, acc, false, false);
    acc = __builtin_amdgcn_wmma_f32_16x16x4_f32(false, A3, false, B3, (short)0, acc, false, false);

    // Stage 2: h = sum_r q^(15-r) * s_r.
    // D layout: acc[j] = s_j (lanes 0-15) or s_{j+8} (lanes 16-31); all columns equal.
    float t;
    t =      (hi ? q7 : q15) * acc[0];
    t = fmaf((hi ? q6 : q14), acc[1], t);
    t = fmaf((hi ? q5 : q13), acc[2], t);
    t = fmaf((hi ? q4 : q12), acc[3], t);
    t = fmaf((hi ? q3 : q11), acc[4], t);
    t = fmaf((hi ? q2 : q10), acc[5], t);
    t = fmaf((hi ? q1 : q9 ), acc[6], t);
    t = fmaf((hi ? q0 : q8 ), acc[7], t);

    t += __shfl_xor(t, 16, 32);  // add the two half-wave contributions
    return t;
}

// Kernel 1: each block streams a contiguous 4096-element chunk; each thread
// serially folds 16 consecutive elements (the scan's affine map applied in
// order), then the block's 256 partials are combined exactly with WMMA.
// Input is single-use: load non-temporal (TH=NT) to keep it out of L2/WGP$.
__global__ __launch_bounds__(BLK) void scan_partials_kernel(
    const float* __restrict__ x, const float* __restrict__ pa,
    const float* __restrict__ pb, const float* __restrict__ pc,
    float* __restrict__ ws) {
    const float a = pa[0], b = pb[0], c = pc[0];
    const int tid = threadIdx.x;
    const size_t base = (size_t)blockIdx.x * (BLK * PER_THREAD) + (size_t)tid * PER_THREAD;

    // 16 contiguous fp32 as 4x global_load_b128 with non-temporal hint
    const v4f* xv = (const v4f*)(x + base);
    v4f q0 = __builtin_nontemporal_load(xv + 0);
    v4f q1 = __builtin_nontemporal_load(xv + 1);
    v4f q2 = __builtin_nontemporal_load(xv + 2);
    v4f q3 = __builtin_nontemporal_load(xv + 3);

    float h = 0.0f;
    h = fmaf(a, h, fmaf(b, q0.x, c));
    h = fmaf(a, h, fmaf(b, q0.y, c));
    h = fmaf(a, h, fmaf(b, q0.z, c));
    h = fmaf(a, h, fmaf(b, q0.w, c));
    h = fmaf(a, h, fmaf(b, q1.x, c));
    h = fmaf(a, h, fmaf(b, q1.y, c));
    h = fmaf(a, h, fmaf(b, q1.z, c));
    h = fmaf(a, h, fmaf(b, q1.w, c));
    h = fmaf(a, h, fmaf(b, q2.x, c));
    h = fmaf(a, h, fmaf(b, q2.y, c));
    h = fmaf(a, h, fmaf(b, q2.z, c));
    h = fmaf(a, h, fmaf(b, q2.w, c));
    h = fmaf(a, h, fmaf(b, q3.x, c));
    h = fmaf(a, h, fmaf(b, q3.y, c));
    h = fmaf(a, h, fmaf(b, q3.z, c));
    h = fmaf(a, h, fmaf(b, q3.w, c));

    __shared__ float part[BLK];
    part[tid] = h;
    __syncthreads();

    if (tid < 32) {                              // one full wave32, EXEC all-1s
        const float a2 = a * a, a4 = a2 * a2, a8 = a4 * a4, a16 = a8 * a8;
        const float r = geo_combine256(part, a16);  // per-partial gain = a^16
        if (tid == 0) ws[blockIdx.x] = r;
    }
}

// Kernel 2: combine the 4096 ordered block partials (block gain P = a^4096,
// computed by fp32 repeated squaring exactly as the reference's fp32 tree
// would), same WMMA combine, then z = w*h + e.
__global__ __launch_bounds__(BLK) void scan_final_kernel(
    const float* __restrict__ ws, const float* __restrict__ pa,
    const float* __restrict__ pw, const float* __restrict__ pe,
    float* __restrict__ out) {
    const float a = pa[0];
    float P = a;
#pragma unroll
    for (int i = 0; i < 12; ++i) P *= P;         // a^4096

    const int tid = threadIdx.x;
    const float* up = ws + tid * 16;
    float h = 0.0f;
#pragma unroll
    for (int j = 0; j < 16; ++j)
        h = fmaf(P, h, up[j]);                   // fold 16 ordered block partials

    __shared__ float part[BLK];
    part[tid] = h;
    __syncthreads();

    if (tid < 32) {
        float p16 = P;
#pragma unroll
        for (int i = 0; i < 4; ++i) p16 *= p16;  // P^16 = a^65536, per-partial gain
        const float r = geo_combine256(part, p16);
        if (tid == 0) out[0] = fmaf(pw[0], r, pe[0]);
    }
}

extern "C" void kernel_launch(void* const* d_in, const int* in_sizes, int n_in,
                              void* d_out, int out_size, void* d_ws, size_t ws_size,
                              hipStream_t stream) {
    const float* x = (const float*)d_in[0];
    const float* a = (const float*)d_in[1];
    const float* b = (const float*)d_in[2];
    const float* c = (const float*)d_in[3];
    const float* w = (const float*)d_in[4];
    const float* e = (const float*)d_in[5];
    float* ws = (float*)d_ws;                    // NBLK floats = 16 KB scratch

    scan_partials_kernel<<<NBLK, BLK, 0, stream>>>(x, a, b, c, ws);
    scan_final_kernel<<<1, BLK, 0, stream>>>(ws, a, w, e, (float*)d_out);
}